// LinkPredictorBackbone_53197464928915
// MI455X (gfx1250) — compile-verified
//
#include <hip/hip_runtime.h>
#include <hip/hip_bf16.h>

typedef __attribute__((ext_vector_type(2))) float v2f;
typedef __attribute__((ext_vector_type(8))) float v8f;

#define HIDC 128
#define BN_EPS 1e-5f

// ---------------- degree / norm ----------------
__global__ void init_deg_kernel(unsigned* __restrict__ deg, int n) {
    int i = blockIdx.x * blockDim.x + threadIdx.x;
    if (i < n) deg[i] = 1u;  // self loop
}

__global__ void count_deg_kernel(const int* __restrict__ dst, unsigned* __restrict__ deg, int e) {
    int i = blockIdx.x * blockDim.x + threadIdx.x;
    if (i < e) atomicAdd(&deg[dst[i]], 1u);
}

__global__ void dinv_kernel(const unsigned* __restrict__ deg, float* __restrict__ dinv, int n) {
    int i = blockIdx.x * blockDim.x + threadIdx.x;
    if (i < n) dinv[i] = rsqrtf((float)deg[i]);   // deg >= 1 always (self loop)
}

// ---------------- fp32 WMMA GEMM: out[nrows x 128] = A[nrows x 128] * W[128 x 128] ----------------
// W staged once per block into LDS, pair-interleaved: wl[k2*128+col] = {W[2k2][col], W[2k2+1][col]}
// so a B fragment for one WMMA is a single conflict-free ds_load_b64.
// One wave computes a 16-row x 128-col stripe (8 tiles of 16x16), K-loop in steps of 4.
template <bool BIAS, bool RELU>
__global__ __launch_bounds__(256) void gemm128_kernel(const float* __restrict__ A,
                                                      const float* __restrict__ W,
                                                      const float* __restrict__ bias,
                                                      float* __restrict__ out, int nrows) {
    __shared__ v2f wl[64 * HIDC];                  // 64 KB

    // cooperative W load: 2048 units of (2 rows x 4 cols); 8 units per thread
    {
        int t = threadIdx.x;
        #pragma unroll
        for (int u = 0; u < 8; ++u) {
            int unit = t + u * 256;                // 0..2047
            int k2 = unit >> 5;                    // 0..63
            int c4 = (unit & 31) * 4;              // column group of 4
            float4 r0 = *(const float4*)(W + (size_t)(2 * k2)     * HIDC + c4);
            float4 r1 = *(const float4*)(W + (size_t)(2 * k2 + 1) * HIDC + c4);
            float4* dst = (float4*)&wl[k2 * HIDC + c4];     // 32B contiguous, 16B aligned
            dst[0] = make_float4(r0.x, r1.x, r0.y, r1.y);
            dst[1] = make_float4(r0.z, r1.z, r0.w, r1.w);
        }
    }
    __syncthreads();

    int wave = (blockIdx.x * blockDim.x + threadIdx.x) >> 5;
    int lane = threadIdx.x & 31;
    int row0 = wave * 16;
    if (row0 + 16 > nrows) return;                 // whole-wave uniform exit (EXEC all-1 for WMMA)

    const int half = lane >> 4;                    // 0: lanes 0-15, 1: lanes 16-31
    const int l16  = lane & 15;
    const int kOff = half * 2;                     // A lanes 0-15 carry K=0,1 ; lanes 16-31 carry K=2,3
    const float* arow = A + (size_t)(row0 + l16) * HIDC;

    v8f acc[8];
    #pragma unroll
    for (int t = 0; t < 8; ++t) acc[t] = (v8f){0.f,0.f,0.f,0.f,0.f,0.f,0.f,0.f};

    for (int k = 0; k < HIDC; k += 4) {
        v2f a = *(const v2f*)(arow + k + kOff);    // 8B aligned (k+kOff even)
        const v2f* brow = &wl[((k >> 1) + half) * HIDC + l16];
        #pragma unroll
        for (int t = 0; t < 8; ++t) {
            v2f b = brow[t * 16];                  // ds_load_b64, bank-conflict-free
            acc[t] = __builtin_amdgcn_wmma_f32_16x16x4_f32(
                false, a, false, b, (short)0, acc[t], false, false);
        }
    }

    // C/D layout: VGPR r -> row row0 + r + 8*half ; column = t*16 + l16
    const int rowAdd = half * 8;
    #pragma unroll
    for (int t = 0; t < 8; ++t) {
        int col = t * 16 + l16;
        float bv = BIAS ? bias[col] : 0.f;
        #pragma unroll
        for (int r = 0; r < 8; ++r) {
            float v = acc[t][r] + bv;
            if (RELU) v = fmaxf(v, 0.f);
            out[(size_t)(row0 + rowAdd + r) * HIDC + col] = v;
        }
    }
}

// ---------------- agg init: agg[i] = hw[i] * dinv[i]^2 (self-loop term folded in) ----------------
__global__ void agg_init_kernel(const float4* __restrict__ hw, const float* __restrict__ dinv,
                                float4* __restrict__ agg, int n) {
    int idx = blockIdx.x * blockDim.x + threadIdx.x;
    int node = idx >> 5;
    if (node >= n) return;
    int c = idx & 31;
    float di = dinv[node];
    float sw = di * di;
    size_t o = (size_t)node * 32 + c;
    float4 v = hw[o];
    agg[o] = make_float4(v.x * sw, v.y * sw, v.z * sw, v.w * sw);
}

// ---------------- edge scatter: agg[dst] += hw[src] * (dinv[src]*dinv[dst]) ----------------
// One wave per edge; each lane moves one float4 of the 128-float row.
__global__ void scatter_kernel(const int* __restrict__ src, const int* __restrict__ dst,
                               const float* __restrict__ dinv, const float* __restrict__ hw,
                               float* __restrict__ agg, int nedges) {
    int e = (blockIdx.x * blockDim.x + threadIdx.x) >> 5;
    if (e >= nedges) return;
    int lane = threadIdx.x & 31;
    int s = src[e], d = dst[e];
    float nrm = dinv[s] * dinv[d];
    float4 v = ((const float4*)(hw + (size_t)s * HIDC))[lane];
    float* drow = agg + (size_t)d * HIDC + lane * 4;
    unsafeAtomicAdd(drow + 0, v.x * nrm);
    unsafeAtomicAdd(drow + 1, v.y * nrm);
    unsafeAtomicAdd(drow + 2, v.z * nrm);
    unsafeAtomicAdd(drow + 3, v.w * nrm);
}

// ---------------- conv bias + BatchNorm(eval) + ReLU, h overwritten in place ----------------
__global__ void bn_relu_kernel(const float4* __restrict__ agg, const float4* __restrict__ cb,
                               const float4* __restrict__ gam, const float4* __restrict__ bet,
                               const float4* __restrict__ mu,  const float4* __restrict__ var,
                               float4* __restrict__ h, int n) {
    int idx = blockIdx.x * blockDim.x + threadIdx.x;
    int node = idx >> 5;
    if (node >= n) return;
    int c = idx & 31;                           // float4 group: 32 per row
    size_t o = (size_t)node * 32 + c;
    float4 a = agg[o];
    float4 b = cb[c], g = gam[c], bt = bet[c], m = mu[c], v = var[c];
    float4 r;
    r.x = fmaxf((a.x + b.x - m.x) * rsqrtf(v.x + BN_EPS) * g.x + bt.x, 0.f);
    r.y = fmaxf((a.y + b.y - m.y) * rsqrtf(v.y + BN_EPS) * g.y + bt.y, 0.f);
    r.z = fmaxf((a.z + b.z - m.z) * rsqrtf(v.z + BN_EPS) * g.z + bt.z, 0.f);
    r.w = fmaxf((a.w + b.w - m.w) * rsqrtf(v.w + BN_EPS) * g.w + bt.w, 0.f);
    h[o] = r;
}

extern "C" void kernel_launch(void* const* d_in, const int* in_sizes, int n_in,
                              void* d_out, int out_size, void* d_ws, size_t ws_size,
                              hipStream_t stream) {
    const float* x      = (const float*)d_in[0];
    const int*   ei     = (const int*)d_in[1];
    /* d_in[2] = edge_attr: unused by the reference math */
    const float* lin0_w = (const float*)d_in[3];
    const float* lin0_b = (const float*)d_in[4];
    const float* lin1_w = (const float*)d_in[5];
    const float* lin1_b = (const float*)d_in[6];
    const float* conv_w = (const float*)d_in[7];
    const float* conv_b = (const float*)d_in[8];
    const float* bn_g   = (const float*)d_in[9];
    const float* bn_b   = (const float*)d_in[10];
    const float* bn_m   = (const float*)d_in[11];
    const float* bn_v   = (const float*)d_in[12];

    const int N = in_sizes[0] / HIDC;
    const int E = in_sizes[1] / 2;
    const int L = in_sizes[7] / (HIDC * HIDC);

    // workspace carve-out (256B aligned): deg, dinv, h, hw, agg  (~154 MB)
    char* base = (char*)d_ws;
    size_t off = 0;
    auto carve = [&](size_t bytes) {
        void* p = base + off;
        off = (off + bytes + 255) & ~(size_t)255;
        return p;
    };
    unsigned* deg  = (unsigned*)carve((size_t)N * 4);
    float*    dinv = (float*)carve((size_t)N * 4);
    float*    h    = (float*)carve((size_t)N * HIDC * 4);
    float*    hw   = (float*)carve((size_t)N * HIDC * 4);
    float*    agg  = (float*)carve((size_t)N * HIDC * 4);
    (void)ws_size; (void)n_in; (void)out_size;

    const int* src = ei;          // edge_index row 0
    const int* dst = ei + E;      // edge_index row 1

    init_deg_kernel<<<(N + 255) / 256, 256, 0, stream>>>(deg, N);
    count_deg_kernel<<<(E + 255) / 256, 256, 0, stream>>>(dst, deg, E);
    dinv_kernel<<<(N + 255) / 256, 256, 0, stream>>>(deg, dinv, N);

    const int rowTiles   = (N + 15) / 16;           // N=100000 -> 6250 exact
    const int gemmBlocks = (rowTiles + 7) / 8;      // 8 waves per 256-thread block
    const int scatBlocks = (E + 7) / 8;             // 8 edges (waves) per block
    const int rowBlocks  = (N * 32 + 255) / 256;    // node x float4-group kernels

    // input projection + ReLU
    gemm128_kernel<true, true><<<gemmBlocks, 256, 0, stream>>>(x, lin0_w, lin0_b, h, N);

    for (int l = 0; l < L; ++l) {
        gemm128_kernel<false, false><<<gemmBlocks, 256, 0, stream>>>(
            h, conv_w + (size_t)l * HIDC * HIDC, nullptr, hw, N);
        agg_init_kernel<<<rowBlocks, 256, 0, stream>>>(
            (const float4*)hw, dinv, (float4*)agg, N);
        scatter_kernel<<<scatBlocks, 256, 0, stream>>>(src, dst, dinv, hw, agg, E);
        bn_relu_kernel<<<rowBlocks, 256, 0, stream>>>(
            (const float4*)agg,
            (const float4*)(conv_b + (size_t)l * HIDC),
            (const float4*)(bn_g  + (size_t)l * HIDC),
            (const float4*)(bn_b  + (size_t)l * HIDC),
            (const float4*)(bn_m  + (size_t)l * HIDC),
            (const float4*)(bn_v  + (size_t)l * HIDC),
            (float4*)h, N);
    }

    // output projection (bias, no ReLU)
    gemm128_kernel<true, false><<<gemmBlocks, 256, 0, stream>>>(h, lin1_w, lin1_b, (float*)d_out, N);
}